// fused_uvu_TP_exp_opt_extcg_23888608100993
// MI455X (gfx1250) — compile-verified
//
#include <hip/hip_runtime.h>
#include <stdint.h>

// Tensor-product message passing + scatter-sum, tuned for MI455X (gfx1250):
//  - wave32: one wave per edge, lanes cover channels (coalesced everywhere)
//  - weight stream (179 MB, dominant HBM traffic) staged via CDNA5 async
//    global->LDS copies (b64+b32), double-buffered per wave (ASYNCcnt pipeline),
//    marked non-temporal so it does not evict out/in1 from the 192 MB L2
//  - segment-sum via native global_atomic_add_f32 (output resident in L2)

#define MUL0      64
#define MUL1      32
#define IN1_SIZE  160   // 64 + 32*3
#define W_SIZE    224   // 64+64+32+32+32
#define W_BYTES   896
#define OUT_SIZE  480   // 64 + 32 + 3*(64+32+32)
#define WPB       8     // waves per block
#define BLOCK     256

#define INV_SQRT3 0.57735026918962584f
#define INV_SQRT2 0.70710678118654752f

// Copy one 896-byte weight row into LDS with async-DMA:
//   3x b64 (lane covers 8B, 256B/instr) + 1x b32 tail (128B).
// The instruction immediate offset is added to BOTH the global address and the
// LDS destination address (CDNA5 ISA pseudocode), so each
// (lds,voff) pair covers its whole stride pattern via offset:.
// th:TH_LOAD_NT keeps this single-use stream from polluting L2.
__device__ __forceinline__ void async_wrow(uint32_t lds8, uint32_t lds4,
                                           uint32_t voff8, uint32_t voff4,
                                           unsigned long long gbase) {
  asm volatile(
      "global_load_async_to_lds_b64 %0, %2, %4 th:TH_LOAD_NT\n\t"
      "global_load_async_to_lds_b64 %0, %2, %4 offset:256 th:TH_LOAD_NT\n\t"
      "global_load_async_to_lds_b64 %0, %2, %4 offset:512 th:TH_LOAD_NT\n\t"
      "global_load_async_to_lds_b32 %1, %3, %4 offset:768 th:TH_LOAD_NT"
      :
      : "v"(lds8), "v"(lds4), "v"(voff8), "v"(voff4), "s"(gbase)
      : "memory");
}

__global__ __launch_bounds__(BLOCK) void tp_scatter_kernel(
    const float* __restrict__ in1,     // [N_NODES, 160]
    const float* __restrict__ in2,     // [N_EDGES, 4]
    const float* __restrict__ weight,  // [N_EDGES, 224]
    const int* __restrict__ esrc,      // [N_EDGES]
    const int* __restrict__ edst,      // [N_EDGES]
    float* __restrict__ out,           // [N_NODES, 480]
    int nEdges) {
  __shared__ float wbuf[WPB][2][W_SIZE];

  const int lane = threadIdx.x & 31;
  const int wid  = threadIdx.x >> 5;
  const int gw   = blockIdx.x * WPB + wid;
  const int nW   = gridDim.x * WPB;

  // Low 32 bits of a flat shared pointer == LDS byte offset on AMDGPU.
  const uint32_t b0   = (uint32_t)(size_t)(&wbuf[wid][0][0]);
  const uint32_t b1   = (uint32_t)(size_t)(&wbuf[wid][1][0]);
  const uint32_t l8   = (uint32_t)(8 * lane);
  const uint32_t l4   = (uint32_t)(4 * lane);
  const unsigned long long wg = (unsigned long long)(size_t)weight;

  int e = gw;
  if (e < nEdges) {
    const uint32_t gb = (uint32_t)e * W_BYTES;
    async_wrow(b0 + l8, b0 + l4, gb + l8, gb + l4, wg);
  }

  int cur = 0;
  for (; e < nEdges; e += nW) {
    const int en = e + nW;
    const bool hasNext = en < nEdges;
    if (hasNext) {
      const uint32_t bb = cur ? b0 : b1;
      const uint32_t gb = (uint32_t)en * W_BYTES;
      async_wrow(bb + l8, bb + l4, gb + l8, gb + l4, wg);
      asm volatile("s_wait_asynccnt 0x4" ::: "memory");  // current row landed
    } else {
      asm volatile("s_wait_asynccnt 0x0" ::: "memory");
    }

    const float* __restrict__ wl = &wbuf[wid][cur][0];
    const float w1a = wl[lane];
    const float w1b = wl[32 + lane];
    const float w2a = wl[64 + lane];
    const float w2b = wl[96 + lane];
    const float w3  = wl[128 + lane];
    const float w4  = wl[160 + lane];
    const float w5  = wl[192 + lane];

    const int s = esrc[e];
    const int d = edst[e];

    // Gathered node features: lanes cover 2 scalar channels + one vec3 channel.
    const float* __restrict__ xr = in1 + (size_t)s * IN1_SIZE;
    const float s1a = xr[lane];
    const float s1b = xr[32 + lane];
    const int   v3  = 64 + 3 * lane;
    const float vx = xr[v3 + 0];
    const float vy = xr[v3 + 1];
    const float vz = xr[v3 + 2];

    const float4 y = *(const float4*)(in2 + (size_t)e * 4);
    const float y0 = y.x, y1x = y.y, y1y = y.z, y1z = y.w;

    // p1 = w1*s1*y0
    const float p1a = w1a * s1a * y0;
    const float p1b = w1b * s1b * y0;
    // p4 = w4 * dot(v1, y1) / sqrt(3)
    const float p4 = w4 * (vx * y1x + vy * y1y + vz * y1z) * INV_SQRT3;
    // p2 = (w2*s1) outer y1
    const float w2sa = w2a * s1a;
    const float w2sb = w2b * s1b;
    // p3 = w3 * v1 * y0
    const float w3y0 = w3 * y0;
    // p5 = w5 * cross(v1, y1) / sqrt(2)
    const float k5 = w5 * INV_SQRT2;
    const float cx = vy * y1z - vz * y1y;
    const float cy = vz * y1x - vx * y1z;
    const float cz = vx * y1y - vy * y1x;

    float* __restrict__ ob = out + (size_t)d * OUT_SIZE;
    const int t3 = 3 * lane;
    // layout: [p1:0..63][p4:64..95][p2:96..287][p3:288..383][p5:384..479]
    unsafeAtomicAdd(ob + lane, p1a);
    unsafeAtomicAdd(ob + 32 + lane, p1b);
    unsafeAtomicAdd(ob + 64 + lane, p4);
    unsafeAtomicAdd(ob + 96 + t3 + 0, w2sa * y1x);
    unsafeAtomicAdd(ob + 96 + t3 + 1, w2sa * y1y);
    unsafeAtomicAdd(ob + 96 + t3 + 2, w2sa * y1z);
    unsafeAtomicAdd(ob + 192 + t3 + 0, w2sb * y1x);
    unsafeAtomicAdd(ob + 192 + t3 + 1, w2sb * y1y);
    unsafeAtomicAdd(ob + 192 + t3 + 2, w2sb * y1z);
    unsafeAtomicAdd(ob + 288 + t3 + 0, w3y0 * vx);
    unsafeAtomicAdd(ob + 288 + t3 + 1, w3y0 * vy);
    unsafeAtomicAdd(ob + 288 + t3 + 2, w3y0 * vz);
    unsafeAtomicAdd(ob + 384 + t3 + 0, k5 * cx);
    unsafeAtomicAdd(ob + 384 + t3 + 1, k5 * cy);
    unsafeAtomicAdd(ob + 384 + t3 + 2, k5 * cz);

    cur ^= 1;
  }
}

extern "C" void kernel_launch(void* const* d_in, const int* in_sizes, int n_in,
                              void* d_out, int out_size, void* d_ws, size_t ws_size,
                              hipStream_t stream) {
  const float* in1    = (const float*)d_in[0];
  const float* in2    = (const float*)d_in[1];
  const float* weight = (const float*)d_in[2];
  const int*   esrc   = (const int*)d_in[3];
  const int*   edst   = (const int*)d_in[4];
  float*       out    = (float*)d_out;

  const int nEdges = in_sizes[3];  // per_edge_src element count

  // Atomic-accumulated output: must start from zero every call.
  hipMemsetAsync(d_out, 0, (size_t)out_size * sizeof(float), stream);

  int blocks = 1024;  // 8192 waves -> ~24 edges/wave: good async pipeline depth
  const int maxBlocks = (nEdges + WPB - 1) / WPB;
  if (blocks > maxBlocks) blocks = maxBlocks;
  if (blocks < 1) blocks = 1;

  tp_scatter_kernel<<<blocks, BLOCK, 0, stream>>>(in1, in2, weight, esrc, edst,
                                                  out, nEdges);
}